// BinaryPositionEmbedding_53077205844631
// MI455X (gfx1250) — compile-verified
//
#include <hip/hip_runtime.h>

typedef float v2f __attribute__((ext_vector_type(2)));
typedef float v8f __attribute__((ext_vector_type(8)));

#define D_MODEL   1024
#define N_BITS    13
#define M_TOTAL   32768            // 4 * 8192 flattened positions
#define M_TILES   (M_TOTAL / 16)   // 2048
#define N_TILES   (D_MODEL / 16)   // 64
#define MT_PER_WAVE 4
#define WAVES_PER_BLOCK 8
#define M_BLOCKS  (M_TILES / (MT_PER_WAVE * WAVES_PER_BLOCK))  // 64

__global__ __launch_bounds__(256)
void BinaryPositionEmbedding_wmma_kernel(const int* __restrict__ x,
                                         const float* __restrict__ w,
                                         float* __restrict__ out) {
    const int lane   = threadIdx.x & 31;   // wave32
    const int wave   = threadIdx.x >> 5;
    const int halfId = lane >> 4;          // 0: lanes 0-15, 1: lanes 16-31
    const int laneM  = lane & 15;

    const int nTile  = blockIdx.x & (N_TILES - 1);
    const int mBlock = blockIdx.x >> 6;    // / N_TILES (=64)
    const int n0     = nTile * 16;

    // ---- Hoisted B: 16-wide column slice of W, K padded 13 -> 16 with zeros.
    // 32-bit B 4x16 layout per K-chunk c (rows 4c..4c+3):
    //   VGPR0: lanes 0-15 -> K=4c+0, lanes 16-31 -> K=4c+2
    //   VGPR1: lanes 0-15 -> K=4c+1, lanes 16-31 -> K=4c+3
    v2f B[4];
#pragma unroll
    for (int c = 0; c < 4; ++c) {
        const int k0 = 4 * c + 2 * halfId;
        float b0 = 0.0f, b1 = 0.0f;
        if (k0 < N_BITS)     b0 = w[k0 * D_MODEL + n0 + laneM];
        if (k0 + 1 < N_BITS) b1 = w[(k0 + 1) * D_MODEL + n0 + laneM];
        B[c].x = b0;
        B[c].y = b1;
    }

    const int mTileBase = (mBlock * WAVES_PER_BLOCK + wave) * MT_PER_WAVE;

#pragma unroll
    for (int t = 0; t < MT_PER_WAVE; ++t) {
        const int m0 = (mTileBase + t) * 16;

        // 16 positions for this M-tile; both half-waves fetch the same 16.
        const int xm = x[m0 + laneM];

        v8f acc = {};  // zero accumulator
#pragma unroll
        for (int c = 0; c < 4; ++c) {
            const int k0 = 4 * c + 2 * halfId;
            // A 16x4 f32 layout: lanes = M rows; VGPR0 = K{4c, 4c+2},
            // VGPR1 = K{4c+1, 4c+3}. Bits >= 13 are 0 because x < 8192.
            v2f a;
            a.x = (float)((xm >> k0) & 1);
            a.y = (float)((xm >> (k0 + 1)) & 1);
            // D = A*B + C, exact f32 matrix op (8 args: neg_a, A, neg_b, B,
            // c_mod, C, reuse_a, reuse_b)
            acc = __builtin_amdgcn_wmma_f32_16x16x4_f32(
                false, a, false, B[c], (short)0, acc, false, false);
        }

        // D layout: VGPR v -> row m0+v (lanes 0-15) / m0+8+v (lanes 16-31),
        // column n0 + (lane & 15). Each store writes two 64B row segments.
        float* o = out + (size_t)(m0 + halfId * 8) * D_MODEL + n0 + laneM;
#pragma unroll
        for (int v = 0; v < 8; ++v) {
            o[(size_t)v * D_MODEL] = acc[v];
        }
    }
}

extern "C" void kernel_launch(void* const* d_in, const int* in_sizes, int n_in,
                              void* d_out, int out_size, void* d_ws, size_t ws_size,
                              hipStream_t stream) {
    const int*   x   = (const int*)d_in[0];    // (4, 8192) int32
    const float* w   = (const float*)d_in[1];  // (13, 1024) f32
    float*       out = (float*)d_out;          // (4, 8192, 1024) f32

    dim3 grid(N_TILES * M_BLOCKS);  // 64 * 64 = 4096 blocks
    dim3 block(256);                // 8 wave32 waves
    BinaryPositionEmbedding_wmma_kernel<<<grid, block, 0, stream>>>(x, w, out);
}